// TruncatedConnection_7447473291325
// MI455X (gfx1250) — compile-verified
//
#include <hip/hip_runtime.h>

#define G_DATA  262144
#define G_TRUNC 40000
#define FEAT    128
#define EPS     1e-8f

#define TILE_E  1024     // edges staged in LDS per block
#define BLOCK   256      // 8 waves (wave32)

#define AS1 __attribute__((address_space(1)))
#define AS3 __attribute__((address_space(3)))

typedef __attribute__((ext_vector_type(4))) int v4i;

// ---- CDNA5 async global->LDS staging ---------------------------------------
#if __has_builtin(__builtin_amdgcn_global_load_async_to_lds_b128) && \
    __has_builtin(__builtin_amdgcn_s_wait_asynccnt)
#define HAVE_ASYNC_LDS 1
#else
#define HAVE_ASYNC_LDS 0
#endif

__device__ __forceinline__ AS3 v4i* to_lds_v4(void* p) {
  // generic shared pointer: low 32 bits are the LDS byte address (ISA 10.2)
  return (AS3 v4i*)(unsigned int)(unsigned long long)p;
}
__device__ __forceinline__ AS1 v4i* to_global_v4(const void* p) {
  return (AS1 v4i*)(unsigned long long)p;
}

__device__ __forceinline__ void copy16_g2lds(const void* g, void* l) {
#if HAVE_ASYNC_LDS
  __builtin_amdgcn_global_load_async_to_lds_b128(to_global_v4(g), to_lds_v4(l), 0, 0);
#else
  *(int4*)l = *(const int4*)g;
#endif
}
__device__ __forceinline__ void wait_async_copies() {
#if HAVE_ASYNC_LDS
  __builtin_amdgcn_s_wait_asynccnt(0);
#endif
}

// ---- f32 atomic add (want native global_atomic_add_f32, agent scope) -------
__device__ __forceinline__ void atomic_add_f32(float* p, float v) {
  __hip_atomic_fetch_add(p, v, __ATOMIC_RELAXED, __HIP_MEMORY_SCOPE_AGENT);
}

// ---- Phase 1: scatter weights to build normalizers -------------------------
__global__ __launch_bounds__(BLOCK) void norm_accum_kernel(
    const int* __restrict__ ddst, const float* __restrict__ dw,
    float* __restrict__ nD,
    const int* __restrict__ udst, const float* __restrict__ uw,
    float* __restrict__ nU, int nEdges)
{
  int i = blockIdx.x * BLOCK + threadIdx.x;
  if (i < nEdges) {
    atomic_add_f32(nD + ddst[i], dw[i]);
    atomic_add_f32(nU + udst[i], uw[i]);
  }
}

// ---- Phase 2/4: edge scatter: accum[dst] += w * xin[src] -------------------
// One wave per edge; 32 lanes x float4 = one 512B feature row per edge.
// Edge (src, dst, w) tiles are streamed into LDS via async copies.
__global__ __launch_bounds__(BLOCK) void scatter_kernel(
    const float* __restrict__ xin,   // [Gsrc, FEAT]
    const int*   __restrict__ src,
    const int*   __restrict__ dst,
    const float* __restrict__ w,
    float*       __restrict__ accum, // [Gdst, FEAT]
    int nEdges)
{
  __shared__ __align__(16) int   s_src[TILE_E];
  __shared__ __align__(16) int   s_dst[TILE_E];
  __shared__ __align__(16) float s_w[TILE_E];

  const int tileBase = blockIdx.x * TILE_E;
  const int t = threadIdx.x;

  // stage the edge tile: each of 256 threads moves 16B per array (4KB/array)
  {
    const int o = t * 4;
    if (tileBase + o + 3 < nEdges) {
      copy16_g2lds(src + tileBase + o, &s_src[o]);
      copy16_g2lds(dst + tileBase + o, &s_dst[o]);
      copy16_g2lds(w   + tileBase + o, &s_w[o]);
    } else {
      for (int k = 0; k < 4; ++k) {
        int e = tileBase + o + k;
        if (e < nEdges) { s_src[o+k] = src[e]; s_dst[o+k] = dst[e]; s_w[o+k] = w[e]; }
      }
    }
  }
  wait_async_copies();
  __syncthreads();

  const int wave = t >> 5;
  const int lane = t & 31;
  const int rem  = nEdges - tileBase;
  const int lim  = rem < TILE_E ? rem : TILE_E;

  for (int k = wave; k < lim; k += BLOCK / 32) {
    const int   s  = s_src[k];        // wave-uniform LDS broadcast
    const int   d  = s_dst[k];
    const float wt = s_w[k];
    const float4 v = ((const float4*)(xin + (size_t)s * FEAT))[lane];
    float* ap = accum + (size_t)d * FEAT + (size_t)lane * 4;
    atomic_add_f32(ap + 0, v.x * wt);
    atomic_add_f32(ap + 1, v.y * wt);
    atomic_add_f32(ap + 2, v.z * wt);
    atomic_add_f32(ap + 3, v.w * wt);
  }
}

// ---- Phase 3/5: row normalization data[g,:] /= max(norm[g], EPS) -----------
__global__ __launch_bounds__(BLOCK) void normalize_kernel(
    float* __restrict__ data, const float* __restrict__ norm, int nNodes)
{
  int i = blockIdx.x * BLOCK + threadIdx.x;   // over nNodes * FEAT/4
  int total = nNodes * (FEAT / 4);
  if (i < total) {
    int node = i / (FEAT / 4);
    float r = 1.0f / fmaxf(norm[node], EPS);
    float4 v = ((const float4*)data)[i];
    v.x *= r; v.y *= r; v.z *= r; v.w *= r;
    ((float4*)data)[i] = v;
  }
}

extern "C" void kernel_launch(void* const* d_in, const int* in_sizes, int n_in,
                              void* d_out, int out_size, void* d_ws, size_t ws_size,
                              hipStream_t stream) {
  const float* x        = (const float*)d_in[0];
  const int*   down_src = (const int*)  d_in[1];
  const int*   down_dst = (const int*)  d_in[2];
  const float* down_w   = (const float*)d_in[3];
  const int*   up_src   = (const int*)  d_in[4];
  const int*   up_dst   = (const int*)  d_in[5];
  const float* up_w     = (const float*)d_in[6];
  float*       out      = (float*)d_out;

  const int E = in_sizes[1];                                  // 1048576
  // x has shape (1,2,1,G_DATA,FEAT); x[:, -1] is the last G_DATA*FEAT block
  const float* x1 = x + ((size_t)in_sizes[0] - (size_t)G_DATA * FEAT);

  // workspace: xc accumulator, then the two normalizers (all 16B aligned)
  float* xc        = (float*)d_ws;                            // G_TRUNC*FEAT
  float* norm_down = xc + (size_t)G_TRUNC * FEAT;             // G_TRUNC
  float* norm_up   = norm_down + G_TRUNC;                     // G_DATA

  const size_t ws_floats = (size_t)G_TRUNC * FEAT + G_TRUNC + G_DATA;
  (void)hipMemsetAsync(d_ws, 0, ws_floats * sizeof(float), stream);
  (void)hipMemsetAsync(d_out, 0, (size_t)G_DATA * FEAT * sizeof(float), stream);

  // 1) normalizers
  norm_accum_kernel<<<(E + BLOCK - 1) / BLOCK, BLOCK, 0, stream>>>(
      down_dst, down_w, norm_down, up_dst, up_w, norm_up, E);

  // 2) down-projection scatter into xc
  scatter_kernel<<<(E + TILE_E - 1) / TILE_E, BLOCK, 0, stream>>>(
      x1, down_src, down_dst, down_w, xc, E);

  // 3) normalize xc
  normalize_kernel<<<(G_TRUNC * (FEAT / 4) + BLOCK - 1) / BLOCK, BLOCK, 0, stream>>>(
      xc, norm_down, G_TRUNC);

  // 4) up-projection scatter into out
  scatter_kernel<<<(E + TILE_E - 1) / TILE_E, BLOCK, 0, stream>>>(
      xc, up_src, up_dst, up_w, out, E);

  // 5) normalize out
  normalize_kernel<<<(G_DATA * (FEAT / 4) + BLOCK - 1) / BLOCK, BLOCK, 0, stream>>>(
      out, norm_up, G_DATA);
}